// Corr_43679817400961
// MI455X (gfx1250) — compile-verified
//
#include <hip/hip_runtime.h>

typedef __attribute__((ext_vector_type(2))) float v2f;
typedef __attribute__((ext_vector_type(8))) float v8f;

#define BATCH 4
#define CHN   64
#define HH    256
#define WW    512
#define DD    40
#define NCH   81           // 2*D+1 output channels
#define LPAD  68           // LDS row stride (floats): banks (4w+c)%64 -> conflict-free b64 frags
#define TPAD  18           // per-wave 16x16 C-tile transpose stride: halves hit disjoint bank sets
#define HW    (HH * WW)

#if __has_builtin(__builtin_amdgcn_global_load_async_to_lds_b32)
#define USE_ASYNC 1
#else
#define USE_ASYNC 0
#endif

#if USE_ASYNC
__device__ __forceinline__ void async_cp_b32(const float* g, const float* l) {
  // AS casts via integer round-trip: low 32 bits of a generic LDS pointer are
  // the DS byte address; global generic == AS1 numerically.
  __builtin_amdgcn_global_load_async_to_lds_b32(
      (__attribute__((address_space(1))) int*)(unsigned long long)(uintptr_t)g,
      (__attribute__((address_space(3))) int*)(uintptr_t)(unsigned)(uintptr_t)l,
      0, 0);
}
#endif

__global__ __launch_bounds__(256)
void corr_wmma_kernel(const float* __restrict__ fL,
                      const float* __restrict__ fR,
                      float* __restrict__ out) {
  extern __shared__ float smem[];
  float* sL    = smem;                    // [512][LPAD]
  float* sR    = smem + WW * LPAD;        // [512][LPAD]
  float* stile = smem + 2 * WW * LPAD;    // [8 waves][16*TPAD]

  const int tid  = threadIdx.x;
  const int wave = tid >> 5;
  const int lane = tid & 31;
  const int half = lane >> 4;             // 0: lanes 0-15, 1: lanes 16-31
  const int lm   = lane & 15;

  const int bh = blockIdx.x;              // b*HH + h
  const int b  = bh / HH;
  const int h  = bh % HH;

  // ---- stage one (b,h) row of fL/fR into LDS, transposed to [w][c] ----
  const size_t inBase = (size_t)b * CHN * HW + (size_t)h * WW;
  for (int w = tid; w < WW; w += 256) {            // lanes -> consecutive w: coalesced reads
#pragma clang loop unroll_count(8)
    for (int c = 0; c < CHN; ++c) {
      size_t g = inBase + (size_t)c * HW + (size_t)w;
#if USE_ASYNC
      async_cp_b32(&fL[g], &sL[w * LPAD + c]);     // direct HBM->LDS scatter, no VGPR round-trip
      async_cp_b32(&fR[g], &sR[w * LPAD + c]);
#else
      sL[w * LPAD + c] = fL[g];
      sR[w * LPAD + c] = fR[g];
#endif
    }
  }

  // ---- zero-fill padded output regions for this (b,h) (d_out is poisoned) ----
  const size_t outBH = (size_t)b * NCH * HW + (size_t)h * WW;
#pragma clang loop unroll(disable)
  for (int c = 1; c < DD; ++c) {                   // left edge of shifted channels (c <= 39 < 256)
    if (tid < c) {
      out[outBH + (size_t)c * HW + tid]         = 0.0f;
      out[outBH + (size_t)(NCH - c) * HW + tid] = 0.0f;
    }
  }
#pragma clang loop unroll(disable)
  for (int w = tid; w < WW; w += 256) {            // channels D and D+1 are all zero
    out[outBH + (size_t)DD * HW + w]       = 0.0f;
    out[outBH + (size_t)(DD + 1) * HW + w] = 0.0f;
  }

#if USE_ASYNC
#if __has_builtin(__builtin_amdgcn_s_wait_asynccnt)
  __builtin_amdgcn_s_wait_asynccnt(0);
#else
  asm volatile("s_wait_asynccnt 0" ::: "memory");
#endif
#endif
  __syncthreads();

  float* mytile = stile + wave * (16 * TPAD);

  // ---- banded GEMM: G = fL_row^T * fR_row, tiles with |tl-tr| <= 3 ----
#pragma clang loop unroll(disable)
  for (int tl = wave; tl < 32; tl += 8) {
    // A tile (16 wl x 64 c) -> 32 VGPRs, reused across all 7 tr neighbors.
    // A frag layout (16x4 f32): lane<16: v0=K0,v1=K1 ; lane>=16: v0=K2,v1=K3
    v2f areg[16];
    const float* aptr = &sL[(tl * 16 + lm) * LPAD + 2 * half];
#pragma unroll
    for (int k = 0; k < 16; ++k)
      areg[k] = *(const v2f*)(aptr + 4 * k);       // ds_load_2addr_b64, conflict-free

#pragma clang loop unroll(disable)
    for (int j = 0; j < 7; ++j) {
      int tr = tl + j - 3;                          // wave-uniform -> EXEC full for WMMA
      if (tr < 0 || tr >= 32) continue;
      const int Delta = 16 * (tl - tr);

      const float* bptr = &sR[(tr * 16 + lm) * LPAD + 2 * half];
      v8f acc = {0.f, 0.f, 0.f, 0.f, 0.f, 0.f, 0.f, 0.f};
#pragma unroll
      for (int k = 0; k < 16; ++k) {
        v2f bfr = *(const v2f*)(bptr + 4 * k);
        acc = __builtin_amdgcn_wmma_f32_16x16x4_f32(
            /*neg_a=*/false, areg[k], /*neg_b=*/false, bfr,
            /*c_mod=*/(short)0, acc, /*reuse_a=*/false, /*reuse_b=*/false);
      }

      // C layout: VGPR v holds M=v (lanes 0-15) / M=v+8 (lanes 16-31), N=lm.
      // Transpose through per-wave LDS so each global store instruction writes
      // consecutive w within one channel (diagonal-major, coalesced).
#pragma unroll
      for (int v = 0; v < 8; ++v)
        mytile[(v + 8 * half) * TPAD + lm] = acc[v];

#pragma clang loop unroll(disable)
      for (int r = 0; r < 16; ++r) {
        int delta = half ? (r + 1) : (r - 15);      // two diagonals per round
        int M = lm + delta;                          // N = lm
        int d = Delta + delta;                       // wl - wr
        bool valid = (delta <= 15) && (M >= 0) && (M <= 15) && (d > -DD) && (d < DD);
        if (valid) {
          float val = mytile[M * TPAD + lm];
          // branchless channel/position select: d>=0 -> (ch d, w=wl); d<0 -> (ch NCH+d, w=wr)
          size_t off = (d >= 0)
                           ? ((size_t)d * HW + (size_t)(tl * 16 + M))
                           : ((size_t)(NCH + d) * HW + (size_t)(tr * 16 + lm));
          out[outBH + off] = val;
        }
      }
    }
  }
}

extern "C" void kernel_launch(void* const* d_in, const int* in_sizes, int n_in,
                              void* d_out, int out_size, void* d_ws, size_t ws_size,
                              hipStream_t stream) {
  (void)in_sizes; (void)n_in; (void)out_size; (void)d_ws; (void)ws_size;
  const float* fL = (const float*)d_in[0];
  const float* fR = (const float*)d_in[1];
  float* out = (float*)d_out;

  const size_t shmem =
      (size_t)(2 * WW * LPAD + 8 * 16 * TPAD) * sizeof(float);  // ~281 KB of 320 KB/WGP
  (void)hipFuncSetAttribute((const void*)corr_wmma_kernel,
                            hipFuncAttributeMaxDynamicSharedMemorySize,
                            (int)shmem);

  corr_wmma_kernel<<<dim3(BATCH * HH), dim3(256), shmem, stream>>>(fL, fR, out);
}